// SimpleMemoryCell_65670049956484
// MI455X (gfx1250) — compile-verified
//
#include <hip/hip_runtime.h>

// Problem constants
#define NB   2048   // batch
#define NU   256    // units
#define NM   64     // memory slots

typedef __attribute__((ext_vector_type(2))) float v2f;
typedef __attribute__((ext_vector_type(8))) float v8f;
typedef __attribute__((ext_vector_type(4))) float f4;

// D = A(16x4) * B(4x16) + C(16x16), all f32, wave32.
// Layouts (per CDNA5 ISA 7.12.2):
//   A: lane L holds row m = L&15; element v holds K = 2*(L>>4) + v  -> contiguous float2 per lane
//   B: element v holds K = 2*(L>>4) + v, col n = L&15
//   C/D: element v holds row M = v + 8*(L>>4), col N = L&15
#define WMMA_F32(a, b, c) \
  __builtin_amdgcn_wmma_f32_16x16x4_f32(false, (a), false, (b), (short)0, (c), false, false)

// ---------------------------------------------------------------------------
// Kernel 1: at = exp(keys0 @ h_t^T)  (64 x 2048), plus column sums.
// One wave per 16-column slab; covers the full M=64 rows (4 stacked tiles),
// so the column sum is local to the wave (one cross-half shuffle).
// ---------------------------------------------------------------------------
__global__ void __launch_bounds__(128) attn_scores_kernel(
    const float* __restrict__ inputs,  // (2048, 512); h_t = cols 256..511
    const float* __restrict__ Mkey,    // (2048, 64, 256); keys0 = Mkey[0]
    float* __restrict__ ws_at,         // (64, 2048) exp scores
    float* __restrict__ ws_sum)        // (2048) column sums
{
  const int lane = threadIdx.x & 31;
  const int wave = threadIdx.x >> 5;
  const int half = lane >> 4;
  const int l15  = lane & 15;
  const int cb   = (blockIdx.x * 4 + wave) * 16;  // column (batch) base

  const float* __restrict__ keys0 = Mkey;  // (64, 256)

  v8f acc[4];
  acc[0] = (v8f)0.f; acc[1] = (v8f)0.f; acc[2] = (v8f)0.f; acc[3] = (v8f)0.f;

  const int n = cb + l15;
#pragma unroll 4
  for (int k = 0; k < NU; k += 4) {
    const int kb = k + 2 * half;
    // B fragment: B[kk][n] = h_t[n][kk] = inputs[n*512 + 256 + kk] (contiguous in kk)
    v2f bfrag = *(const v2f*)&inputs[n * (2 * NU) + NU + kb];
#pragma unroll
    for (int mt = 0; mt < 4; ++mt) {
      v2f afrag = *(const v2f*)&keys0[(mt * 16 + l15) * NU + kb];
      acc[mt] = WMMA_F32(afrag, bfrag, acc[mt]);
    }
  }

  // exp + store + per-lane partial column sum
  float psum = 0.f;
#pragma unroll
  for (int mt = 0; mt < 4; ++mt) {
#pragma unroll
    for (int v = 0; v < 8; ++v) {
      float e = __expf(acc[mt][v]);
      psum += e;
      ws_at[(mt * 16 + v + 8 * half) * NB + cb + l15] = e;
    }
  }
  // lanes L and L+16 hold the same column (different row halves)
  float tot = psum + __shfl_xor(psum, 16, 32);
  if (half == 0) ws_sum[cb + l15] = tot;
}

// ---------------------------------------------------------------------------
// Kernel 2: output = atn^T @ vals0  (2048 x 256)
// Quirky divisor from reshape: atn[m,c] = at[m,c] / at_sum[m*32 + c/64]
// One wave per 16x16 output tile; K = 64.
// ---------------------------------------------------------------------------
__global__ void __launch_bounds__(128) attn_out_kernel(
    const float* __restrict__ ws_at,   // (64, 2048)
    const float* __restrict__ ws_sum,  // (2048)
    const float* __restrict__ Mval,    // (2048, 64, 256); vals0 = Mval[0]
    float* __restrict__ out)           // (2048, 256)
{
  const int lane = threadIdx.x & 31;
  const int wave = threadIdx.x >> 5;
  const int half = lane >> 4;
  const int l15  = lane & 15;
  const int tile = blockIdx.x * 4 + wave;       // 2048 tiles
  const int cb   = (tile & 127) * 16;           // batch block
  const int ub   = (tile >> 7) * 16;            // unit block

  const float* __restrict__ vals0 = Mval;  // (64, 256)

  v8f acc = (v8f)0.f;
  const int c = cb + l15;
  const int cd = c >> 6;  // divisor sub-index
#pragma unroll 4
  for (int k = 0; k < NM; k += 4) {
    const int m0 = k + 2 * half;
    // A fragment: atn^T[c][m] with the reshape-divisor applied on load
    v2f af;
    af.x = ws_at[ m0      * NB + c] / ws_sum[ m0      * 32 + cd];
    af.y = ws_at[(m0 + 1) * NB + c] / ws_sum[(m0 + 1) * 32 + cd];
    // B fragment: vals0[m][u]
    v2f bf;
    bf.x = vals0[ m0      * NU + ub + l15];
    bf.y = vals0[(m0 + 1) * NU + ub + l15];
    acc = WMMA_F32(af, bf, acc);
  }
#pragma unroll
  for (int v = 0; v < 8; ++v)
    out[(cb + v + 8 * half) * NU + ub + l15] = acc[v];
}

// ---------------------------------------------------------------------------
// Kernel 3: m_key = e_t @ W_key, m_value = e_t @ W_value, written directly
// into slot 63 of M_key_new / M_value_new. Shares A-fragment loads.
// ---------------------------------------------------------------------------
__global__ void __launch_bounds__(128) update_gemm_kernel(
    const float* __restrict__ inputs,  // (2048, 512); e_t = cols 0..255
    const float* __restrict__ Wkey,    // (256, 256)
    const float* __restrict__ Wval,    // (256, 256)
    float* __restrict__ outMkey,       // (2048, 64, 256)
    float* __restrict__ outMval)       // (2048, 64, 256)
{
  const int lane = threadIdx.x & 31;
  const int wave = threadIdx.x >> 5;
  const int half = lane >> 4;
  const int l15  = lane & 15;
  const int tile = blockIdx.x * 4 + wave;  // 2048 tiles
  const int cb   = (tile & 127) * 16;      // batch block
  const int ub   = (tile >> 7) * 16;       // unit block

  v8f acck = (v8f)0.f;
  v8f accv = (v8f)0.f;
  const int u = ub + l15;
#pragma unroll 4
  for (int k = 0; k < NU; k += 4) {
    const int kb = k + 2 * half;
    // A fragment: e_t[c][k] = inputs[c*512 + k] (contiguous float2)
    v2f af = *(const v2f*)&inputs[(cb + l15) * (2 * NU) + kb];
    v2f bk, bv;
    bk.x = Wkey[ kb      * NU + u];
    bk.y = Wkey[(kb + 1) * NU + u];
    bv.x = Wval[ kb      * NU + u];
    bv.y = Wval[(kb + 1) * NU + u];
    acck = WMMA_F32(af, bk, acck);
    accv = WMMA_F32(af, bv, accv);
  }
#pragma unroll
  for (int v = 0; v < 8; ++v) {
    const int c = cb + v + 8 * half;
    outMkey[(c * NM + 63) * NU + u] = acck[v];
    outMval[(c * NM + 63) * NU + u] = accv[v];
  }
}

// ---------------------------------------------------------------------------
// Kernel 4: memory shift (the bandwidth-dominant op, ~512 MB of traffic).
// M_new[b, r, :] = M_old[b, r+1, :] for r in [0,63). In float4 units, one
// row = 64 float4, one batch = 4096 float4, so src = dst + 64 within a batch.
// Nontemporal b128 load/store: streaming data, don't pollute caches.
// ---------------------------------------------------------------------------
__global__ void __launch_bounds__(256) shift_copy_kernel(
    const f4* __restrict__ srcK, const f4* __restrict__ srcV,
    f4* __restrict__ dstK, f4* __restrict__ dstV, unsigned total4)
{
  unsigned i = blockIdx.x * 256u + threadIdx.x;
  if (i >= total4) return;                 // total4 = 2048*63*64 = 8257536
  unsigned b   = i / 4032u;                // 63 rows * 64 float4
  unsigned rem = i - b * 4032u;
  unsigned d   = b * 4096u + rem;          // dst index; src = dst + 64
  f4 vk = __builtin_nontemporal_load(&srcK[d + 64]);
  f4 vv = __builtin_nontemporal_load(&srcV[d + 64]);
  __builtin_nontemporal_store(vk, &dstK[d]);
  __builtin_nontemporal_store(vv, &dstV[d]);
}

// ---------------------------------------------------------------------------
extern "C" void kernel_launch(void* const* d_in, const int* in_sizes, int n_in,
                              void* d_out, int out_size, void* d_ws, size_t ws_size,
                              hipStream_t stream) {
  const float* inputs = (const float*)d_in[0];  // (2048, 512)
  const float* Mkey   = (const float*)d_in[1];  // (2048, 64, 256)
  const float* Mval   = (const float*)d_in[2];  // (2048, 64, 256)
  const float* Wkey   = (const float*)d_in[3];  // (256, 256)
  const float* Wval   = (const float*)d_in[4];  // (256, 256)

  float* out     = (float*)d_out;                       // (2048, 256)
  float* outMkey = out + (size_t)NB * NU;               // (2048, 64, 256)
  float* outMval = outMkey + (size_t)NB * NM * NU;      // (2048, 64, 256)

  float* ws_at  = (float*)d_ws;          // 64*2048 floats = 512 KB
  float* ws_sum = ws_at + NM * NB;       // 2048 floats

  // 1) exp scores + column sums (128 waves)
  attn_scores_kernel<<<32, 128, 0, stream>>>(inputs, Mkey, ws_at, ws_sum);
  // 2) attention output GEMM (2048 tiles)
  attn_out_kernel<<<512, 128, 0, stream>>>(ws_at, ws_sum, Mval, out);
  // 3) new key/value rows -> slot 63 of the output tensors (2048 tiles)
  update_gemm_kernel<<<512, 128, 0, stream>>>(inputs, Wkey, Wval, outMkey, outMval);
  // 4) bandwidth-dominant shift copy, slots 0..62 (disjoint from kernel 3's writes)
  const unsigned total4 = (unsigned)NB * 63u * (NU / 4);  // 8257536
  shift_copy_kernel<<<(total4 + 255) / 256, 256, 0, stream>>>(
      (const f4*)Mkey, (const f4*)Mval, (f4*)outMkey, (f4*)outMval, total4);
}